// CustomDetectionLoss_10763188044396
// MI455X (gfx1250) — compile-verified
//
#include <hip/hip_runtime.h>
#include <math.h>

#define NUMC   85
#define NA3    6400
#define NA4    1600
#define NANCH  8400
#define TOPKN  50
#define NBATCH 32
#define NGT    640
#define NCLS   80

typedef float v2f __attribute__((ext_vector_type(2)));
typedef float v8f __attribute__((ext_vector_type(8)));

// pred[b, i, ch] for the concatenated (p3|p4|p5) anchor axis, reading only what we need.
__device__ __forceinline__ float pred_at(const float* __restrict__ p3,
                                         const float* __restrict__ p4,
                                         const float* __restrict__ p5,
                                         int b, int i, int ch) {
  if (i < NA3)        return p3[((b * NUMC + ch) * NA3) + i];
  if (i < NA3 + NA4)  return p4[((b * NUMC + ch) * NA4) + (i - NA3)];
  return p5[((b * NUMC + ch) * 400) + (i - NA3 - NA4)];
}

__device__ __forceinline__ float bce(float x, float t) {
  return fmaxf(x, 0.f) - x * t + log1pf(expf(-fabsf(x)));
}

// ---------------- Kernel 1: per-image top-50 of the obj channel ----------------
__global__ void topk_kernel(const float* __restrict__ p3, const float* __restrict__ p4,
                            const float* __restrict__ p5, int* __restrict__ sel) {
  __shared__ float obj[NANCH];
  __shared__ float wv[8];
  __shared__ int   wi[8];
  const int b = blockIdx.x, tid = threadIdx.x;

  // CDNA5 prefetch path (lowers to global_prefetch_b8)
  __builtin_prefetch(&p3[(b * NUMC + 4) * NA3], 0, 3);

  for (int i = tid; i < NANCH; i += 256) obj[i] = pred_at(p3, p4, p5, b, i, 4);
  __syncthreads();

  for (int s = 0; s < TOPKN; ++s) {
    float v = -__builtin_inff();
    int   idx = 0;
    for (int i = tid; i < NANCH; i += 256) {
      float o = obj[i];
      if (o > v) { v = o; idx = i; }   // strict >: first (lowest) index wins ties
    }
    // wave32 shuffle reduction (value desc, index asc on ties)
    for (int off = 16; off; off >>= 1) {
      float ov = __shfl_xor(v, off);
      int   oi = __shfl_xor(idx, off);
      if (ov > v || (ov == v && oi < idx)) { v = ov; idx = oi; }
    }
    const int w = tid >> 5, lane = tid & 31;
    if (lane == 0) { wv[w] = v; wi[w] = idx; }
    __syncthreads();
    if (tid == 0) {
      float bv = wv[0]; int bi = wi[0];
      for (int q = 1; q < 8; ++q)
        if (wv[q] > bv || (wv[q] == bv && wi[q] < bi)) { bv = wv[q]; bi = wi[q]; }
      obj[bi] = -__builtin_inff();
      sel[b * TOPKN + s] = bi;
    }
    __syncthreads();
  }
}

// map (row m, k, iter t) of the WMMA A fragment onto the three arrays to be summed
__device__ __forceinline__ float fragval(const float* bx, const float* ob, const float* cr,
                                         int m, int k, int t) {
  const int e = k * 4 + t;              // 0..15
  if (m < 4)  return bx[m * 16 + e];    // rows 0-3 : 64 slots (50 box terms + zeros)
  if (m < 8)  return ob[(m - 4) * 16 + e]; // rows 4-7 : 64 slots (50 obj terms + zeros)
  return cr[(m - 8) * 16 + e];          // rows 8-15: 128 pre-reduced cls partials
}

// ---------------- Kernel 2: CIoU match + losses per image ----------------
__global__ void loss_kernel(const float* __restrict__ p3, const float* __restrict__ p4,
                            const float* __restrict__ p5, const float* __restrict__ bboxes,
                            const int* __restrict__ cls, const int* __restrict__ bidx,
                            const int* __restrict__ sel, float* __restrict__ acc) {
  __shared__ float sbx[NGT], sby[NGT], sbw[NGT], sbh[NGT];
  __shared__ int   sb[NGT], sc[NGT];
  __shared__ int   sidx[TOPKN], cidx[TOPKN];
  __shared__ float boxterm[64], objterm[64];
  __shared__ float clspart[256], clsred[128];
  __shared__ int   shas;

  const int b = blockIdx.x, tid = threadIdx.x;
  if (tid == 0) shas = 0;
  if (tid < 64) { boxterm[tid] = 0.f; objterm[tid] = 0.f; }
  __syncthreads();

  for (int j = tid; j < NGT; j += 256) {
    sbx[j] = bboxes[4 * j + 0]; sby[j] = bboxes[4 * j + 1];
    sbw[j] = bboxes[4 * j + 2]; sbh[j] = bboxes[4 * j + 3];
    const int bb = bidx[j];
    sb[j] = bb; sc[j] = cls[j];
    if (bb == b) shas = 1;               // benign race: all writers store 1
  }
  if (tid < TOPKN) sidx[tid] = sel[b * TOPKN + tid];
  __syncthreads();

  const int wv = tid >> 5, lane = tid & 31;

  // each wave owns anchors k = wv, wv+8, ...; 32 lanes sweep the 640 GT boxes
  for (int k = wv; k < TOPKN; k += 8) {
    const int i = sidx[k];
    const float b1x = pred_at(p3, p4, p5, b, i, 0);
    const float b1y = pred_at(p3, p4, p5, b, i, 1);
    const float b1w = pred_at(p3, p4, p5, b, i, 2);
    const float b1h = pred_at(p3, p4, p5, b, i, 3);
    const float so  = pred_at(p3, p4, p5, b, i, 4);
    const float b1x1 = b1x - 0.5f * b1w, b1y1 = b1y - 0.5f * b1h;
    const float b1x2 = b1x + 0.5f * b1w, b1y2 = b1y + 0.5f * b1h;
    const float b1area = b1w * b1h;
    const float at1 = atanf(b1w / b1h);

    float mv = -2.f; int mi = 0;
    for (int j = lane; j < NGT; j += 32) {
      float v;
      if (sb[j] == b) {
        const float b2x = sbx[j], b2y = sby[j], b2w = sbw[j], b2h = sbh[j];
        const float b2x1 = b2x - 0.5f * b2w, b2y1 = b2y - 0.5f * b2h;
        const float b2x2 = b2x + 0.5f * b2w, b2y2 = b2y + 0.5f * b2h;
        float iw = fminf(b1x2, b2x2) - fmaxf(b1x1, b2x1); iw = fmaxf(iw, 0.f);
        float ih = fminf(b1y2, b2y2) - fmaxf(b1y1, b2y1); ih = fmaxf(ih, 0.f);
        const float inter = iw * ih;
        const float uni   = b1area + b2w * b2h - inter + 1e-7f;
        const float iou   = inter / uni;
        const float cw  = fmaxf(b1x2, b2x2) - fminf(b1x1, b2x1);
        const float chh = fmaxf(b1y2, b2y2) - fminf(b1y1, b2y1);
        const float c2  = cw * cw + chh * chh + 1e-7f;
        const float dx = b2x - b1x, dy = b2y - b1y;
        const float rho2 = dx * dx + dy * dy;
        const float dat = atanf(b2w / b2h) - at1;
        const float vterm = 0.4052847345693511f * dat * dat;   // 4/pi^2
        const float alpha = vterm / (1.f - iou + vterm + 1e-7f);
        v = iou - (rho2 / c2 + vterm * alpha);
        v = fminf(fmaxf(v, 0.f), 1.f);
      } else {
        v = -1.f;
      }
      if (v > mv) { mv = v; mi = j; }    // strict >: keeps lowest j on ties
    }
    for (int off = 16; off; off >>= 1) {
      float ov = __shfl_xor(mv, off);
      int   oi = __shfl_xor(mi, off);
      if (ov > mv || (ov == mv && oi < mi)) { mv = ov; mi = oi; }
    }
    if (lane == 0) {
      cidx[k]    = mi;
      boxterm[k] = 1.f - mv;
      objterm[k] = bce(so, mv);          // obj_t = stop_gradient(ciou_max)
    }
  }
  __syncthreads();

  // cls BCE: 50*80 = 4000 terms
  float part = 0.f;
  for (int p = tid; p < TOPKN * NCLS; p += 256) {
    const int k = p / NCLS, c = p - k * NCLS;
    const int gtc = sc[cidx[k]];
    const float x = pred_at(p3, p4, p5, b, sidx[k], 5 + c);
    part += bce(x, (c == gtc) ? 1.f : 0.f);
  }
  clspart[tid] = part;
  __syncthreads();
  if (tid < 128) clsred[tid] = clspart[tid] + clspart[tid + 128];
  __syncthreads();

  const float mf = shas ? 1.f : 0.f;

#if defined(__gfx1250__) && __has_builtin(__builtin_amdgcn_wmma_f32_16x16x4_f32)
  // Exact-f32 segmented reduction on the matrix pipe: D = A x ones + C.
  // Row m of D accumulates sum_k A[m][k]; rows 0-3 => box, 4-7 => obj, 8-15 => cls.
  // All 8 waves execute uniformly (EXEC all-ones as WMMA requires); only
  // threads 0 / 16 (wave 0, lanes 0 / 16) commit results.
  v2f ones; ones.x = 1.f; ones.y = 1.f;
  v8f cfrag = {};
  const int m = lane & 15, kp = lane >> 4;   // A 16x4 layout: lanes 0-15 K=0,1; 16-31 K=2,3
  for (int t = 0; t < 4; ++t) {
    v2f a;
    a.x = fragval(boxterm, objterm, clsred, m, 2 * kp + 0, t);
    a.y = fragval(boxterm, objterm, clsred, m, 2 * kp + 1, t);
    cfrag = __builtin_amdgcn_wmma_f32_16x16x4_f32(false, a, false, ones,
                                                  (short)0, cfrag, false, false);
  }
  // D layout: VGPR j, lanes 0-15 = row j (col = lane); lanes 16-31 = row j+8.
  const float s_lo = cfrag[0] + cfrag[1] + cfrag[2] + cfrag[3];
  const float s_hi = cfrag[4] + cfrag[5] + cfrag[6] + cfrag[7];
  if (tid == 0) {   // lane 0, col 0: rows 0-3 (box), rows 4-7 (obj)
    atomicAdd(&acc[0], mf * s_lo * (1.f / TOPKN));
    atomicAdd(&acc[1], mf * s_hi * (1.f / TOPKN));
  }
  if (tid == 16) {  // lane 16, col 0: rows 8-15 (cls)
    atomicAdd(&acc[2], mf * (s_lo + s_hi) * (1.f / (TOPKN * NCLS)));
  }
#else
  if (tid == 0) {
    float sb_ = 0.f, so_ = 0.f, sc_ = 0.f;
    for (int k = 0; k < TOPKN; ++k) { sb_ += boxterm[k]; so_ += objterm[k]; }
    for (int q = 0; q < 128; ++q)    sc_ += clsred[q];
    atomicAdd(&acc[0], mf * sb_ * (1.f / TOPKN));
    atomicAdd(&acc[1], mf * so_ * (1.f / TOPKN));
    atomicAdd(&acc[2], mf * sc_ * (1.f / (TOPKN * NCLS)));
  }
#endif
}

// ---------------- Kernel 3: finalize ----------------
__global__ void finalize_kernel(const float* __restrict__ acc, float* __restrict__ out) {
  if (threadIdx.x == 0 && blockIdx.x == 0) {
    const float lb = acc[0] * (1.f / NBATCH);
    const float lo = acc[1] * (1.f / NBATCH);
    const float lc = acc[2] * (1.f / NBATCH);
    out[0] = 0.05f * lb + 1.0f * lo + 0.5f * lc;
    out[1] = lb;
    out[2] = lo;
    out[3] = lc;
  }
}

extern "C" void kernel_launch(void* const* d_in, const int* in_sizes, int n_in,
                              void* d_out, int out_size, void* d_ws, size_t ws_size,
                              hipStream_t stream) {
  const float* p3     = (const float*)d_in[0];
  const float* p4     = (const float*)d_in[1];
  const float* p5     = (const float*)d_in[2];
  const float* bboxes = (const float*)d_in[3];
  const int*   cls    = (const int*)d_in[4];
  const int*   bidx   = (const int*)d_in[5];

  float* acc = (float*)d_ws;                       // 3 f32 accumulators
  int*   sel = (int*)((char*)d_ws + 64);           // 32 x 50 top-k indices

  hipMemsetAsync(d_ws, 0, 64, stream);             // graph-capture-safe zeroing
  topk_kernel<<<dim3(NBATCH), dim3(256), 0, stream>>>(p3, p4, p5, sel);
  loss_kernel<<<dim3(NBATCH), dim3(256), 0, stream>>>(p3, p4, p5, bboxes, cls, bidx, sel, acc);
  finalize_kernel<<<dim3(1), dim3(32), 0, stream>>>(acc, (float*)d_out);
}